// NRODESolver_33938831573389
// MI455X (gfx1250) — compile-verified
//
#include <hip/hip_runtime.h>
#include <math.h>

#define BS 200000
#define NY 8
#define NTH 256
#define NBLK ((BS + NTH - 1) / NTH)
#define N_STEPS 10

#define TOLX_F   1e-12f
#define TOLFUN_F 1e-6f
#define ALPHA_F  1e-4f
#define MINLAM   0.1f
#define MAXLAM   0.5f
#define MAXITER_I 1000

typedef float v2f __attribute__((ext_vector_type(2)));
typedef float v8f __attribute__((ext_vector_type(8)));

// ---------------- global scalar state (lives at d_ws offset 0) ----------------
struct Glb {
  float lam, lam1, lam2, lam_min;
  int   niter, exitflag;
  int   active, is_newton, recompute, upd, cont;
  float maxRes, maxRatio;
};

// ---------------- workspace layout (byte offsets into d_ws) ----------------
constexpr size_t PA_OFF = 256;                                          // phaseA partials: NBLK x 12 f32
constexpr size_t PR_OFF = PA_OFF + (size_t)NBLK * 12 * sizeof(float);   // resnorm partials: NBLK x 2 f32
constexpr size_t ARR0   = (PR_OFF + (size_t)NBLK * 2 * sizeof(float) + 255) & ~(size_t)255;
constexpr size_t SZ8 = (size_t)BS * NY * sizeof(float);
constexpr size_t SZ1 = (size_t)BS * sizeof(float);
constexpr size_t YOLD_O  = ARR0;            // state: yold
constexpr size_t DY_O    = YOLD_O + SZ8;    // state: dy
constexpr size_t SB_O    = DY_O   + SZ8;    // state: jac basis s = sech^2(u)  (jac = cI - (W S V)^T)
constexpr size_t YTR_O   = SB_O   + SZ8;    // trial y
constexpr size_t FTR_O   = YTR_O  + SZ8;    // trial F
constexpr size_t DYTR_O  = FTR_O  + SZ8;    // trial dy
constexpr size_t SBTR_O  = DYTR_O + SZ8;    // trial jac basis
constexpr size_t F2_O    = SBTR_O + SZ8;    // state: f2
constexpr size_t FOLD_O  = F2_O   + SZ1;    // state: fold
constexpr size_t SLOPE_O = FOLD_O + SZ1;    // state: slope
constexpr size_t SLTR_O  = SLOPE_O+ SZ1;    // trial slope
constexpr size_t RN_O    = SLTR_O + SZ1;    // state: resnorm
constexpr size_t RN0_O   = RN_O   + SZ1;    // state: resnorm0

__device__ inline float* wsf(void* ws, size_t off) { return (float*)((char*)ws + off); }

__device__ inline float sdiv(float n, float d) { return (fabsf(d) > 1e-30f) ? (n / d) : 0.0f; }
__device__ inline float finf(float x) { return (fabsf(x) < INFINITY) ? 1.0f : 0.0f; }

// Block-wide tree reduction (blockDim.x == NTH), all threads must participate.
__device__ inline float block_red(float v, int isMin, float* rbuf) {
  int t = threadIdx.x;
  rbuf[t] = v; __syncthreads();
  for (int s = NTH / 2; s > 0; s >>= 1) {
    if (t < s) rbuf[t] = isMin ? fminf(rbuf[t], rbuf[t + s]) : fmaxf(rbuf[t], rbuf[t + s]);
    __syncthreads();
  }
  float r = rbuf[0]; __syncthreads();
  return r;
}

// ---------------- f32 WMMA: D(16x16) += A(16x4) x B(4x16) ----------------
__device__ inline v8f wmma4(v2f a, v2f b, v8f c) {
  // (neg_a, A, neg_b, B, c_mod, C, reuse_a, reuse_b)
  return __builtin_amdgcn_wmma_f32_16x16x4_f32(false, a, false, b, (short)0, c, false, false);
}

// Per-wave batched vec(8) @ M(8x8, zero-padded to 8x16 in LDS) for 32 samples.
// stw = this wave's 32x8 LDS staging; lM = [k][n] row-major 8x16.
// Fragment layouts per CDNA5 ISA 7.12.2 (32-bit A 16x4, 32-bit C/D 16x16).
__device__ void wave_matvec8(float* stw, const float* lM, const float* in8, float* out8) {
  const int  lane = threadIdx.x & 31;
  const int  l15  = lane & 15;
  const bool hi   = lane >= 16;
#pragma unroll
  for (int k = 0; k < 8; k++) stw[lane * 8 + k] = in8[k];
  __syncthreads();
  v8f acc0 = {}; v8f acc1 = {};
#pragma unroll
  for (int c = 0; c < 2; c++) {            // K = 8 as two K=4 chunks
    const int k0 = c * 4 + (hi ? 2 : 0);   // A: low lanes hold K{0,1}, high lanes K{2,3}
    v2f a0, a1, b;
    a0.x = stw[l15 * 8 + k0];        a0.y = stw[l15 * 8 + k0 + 1];        // tile0: samples 0..15
    a1.x = stw[(16 + l15) * 8 + k0]; a1.y = stw[(16 + l15) * 8 + k0 + 1]; // tile1: samples 16..31
    b.x  = lM[k0 * 16 + l15];        b.y  = lM[(k0 + 1) * 16 + l15];      // B rows K, cols N=l15
    acc0 = wmma4(a0, b, acc0);
    acc1 = wmma4(a1, b, acc1);
  }
  __syncthreads();                         // frag loads consumed -> safe to overwrite stage
  if (l15 < 8) {                           // D: lane holds column N=l15; VGPR r = row (hi: +8)
    const int r0 = hi ? 8 : 0;
#pragma unroll
    for (int r = 0; r < 8; r++) {
      stw[(r0 + r) * 8 + l15]      = acc0[r];
      stw[(16 + r0 + r) * 8 + l15] = acc1[r];
    }
  }
  __syncthreads();
#pragma unroll
  for (int k = 0; k < 8; k++) out8[k] = stw[lane * 8 + k];
  __syncthreads();
}

__device__ void load_WV(const float* W, const float* V, float* lW, float* lV) {
  for (int i = threadIdx.x; i < 128; i += NTH) {
    int r = i >> 4, n = i & 15;
    lW[i] = (n < 8) ? W[r * 8 + n] : 0.0f;
    lV[i] = (n < 8) ? V[r * 8 + n] : 0.0f;
  }
  __syncthreads();
}

// ---------------- kernels ----------------
__global__ void k_init_glb(void* ws) {
  Glb* g = (Glb*)ws;
  g->lam = 1.f; g->lam1 = 1.f; g->lam2 = 0.5f; g->lam_min = 0.f;
  g->niter = 0; g->exitflag = 1;
  g->active = 0; g->is_newton = 0; g->recompute = 0; g->upd = 0; g->cont = 0;
  g->maxRes = 0.f; g->maxRatio = 0.f;
}

__global__ __launch_bounds__(NTH) void k_init(const float* y0, const float* W, const float* V,
                                              const float* tp, float* out, void* ws) {
  __shared__ float lW[128], lV[128];
  __shared__ float stage[NTH / 32][32 * 8];
  __shared__ float rbuf[NTH];
  load_WV(W, V, lW, lV);
  const int  idx   = blockIdx.x * NTH + threadIdx.x;
  const bool valid = idx < BS;
  const int  si    = valid ? idx : (BS - 1);
  const float tt   = tp[0];
  float y[8];
#pragma unroll
  for (int k = 0; k < 8; k++) y[k] = y0[(size_t)si * 8 + k];
  float* stw = stage[threadIdx.x >> 5];
  float u[8], th[8], z[8];
  wave_matvec8(stw, lW, y, u);
#pragma unroll
  for (int k = 0; k < 8; k++) th[k] = tanhf(u[k]);
  wave_matvec8(stw, lV, th, z);
  const float st = 0.01f * sinf(tt);
  float F[8], sb[8], fobj = 0.f, rn = 0.f;
#pragma unroll
  for (int k = 0; k < 8; k++) {
    F[k]  = -z[k] + 0.1f * y[k] - st;   // (y0-y0)/dt == 0
    sb[k] = 1.0f - th[k] * th[k];
    fobj += F[k] * F[k];
    rn = fmaxf(rn, fabsf(F[k]));
  }
  if (valid) {
#pragma unroll
    for (int k = 0; k < 8; k++) {
      out[(size_t)si * 8 + k]                    = y[k];
      out[(size_t)BS * NY + (size_t)si * 8 + k]  = F[k];
      wsf(ws, YOLD_O)[(size_t)si * 8 + k] = y[k];
      wsf(ws, DY_O)[(size_t)si * 8 + k]   = 0.0f;
      wsf(ws, SB_O)[(size_t)si * 8 + k]   = sb[k];
    }
    wsf(ws, F2_O)[si] = fobj;  wsf(ws, FOLD_O)[si] = fobj;
    wsf(ws, SLOPE_O)[si] = 0.0f;
    wsf(ws, RN_O)[si] = rn;    wsf(ws, RN0_O)[si] = 100.0f * rn;
  }
  float ratio = sdiv(rn, 100.0f * rn);
  float r0 = block_red(valid ? rn : -INFINITY, 0, rbuf);
  float r1 = block_red(valid ? ratio : -INFINITY, 0, rbuf);
  if (threadIdx.x == 0) {
    wsf(ws, PR_OFF)[blockIdx.x * 2 + 0] = r0;
    wsf(ws, PR_OFF)[blockIdx.x * 2 + 1] = r1;
  }
}

__global__ __launch_bounds__(NTH) void k_reduceR(void* ws) {
  __shared__ float rbuf[NTH];
  Glb* g = (Glb*)ws;
  float* pr = wsf(ws, PR_OFF);
  float m0 = -INFINITY, m1 = -INFINITY;
  for (int i = threadIdx.x; i < NBLK; i += NTH) {
    m0 = fmaxf(m0, pr[i * 2 + 0]);
    m1 = fmaxf(m1, pr[i * 2 + 1]);
  }
  m0 = block_red(m0, 0, rbuf);
  m1 = block_red(m1, 0, rbuf);
  if (threadIdx.x == 0) {
    g->maxRes = m0; g->maxRatio = m1;
    g->active = (((m0 > TOLFUN_F) || (g->lam < 1.0f)) && (g->exitflag >= 0) && (g->niter <= MAXITER_I)) ? 1 : 0;
    g->is_newton = (g->lam == 1.0f) ? 1 : 0;
    g->recompute = (m1 > 0.2f) ? 1 : 0;
  }
}

__global__ __launch_bounds__(NTH) void k_phaseA(const float* y0, const float* W, const float* V,
                                                const float* dtp, const float* tp,
                                                const float* out, void* ws) {
  Glb* g = (Glb*)ws;
  if (!g->active) return;                         // grid-uniform
  __shared__ float lW[128], lV[128];
  __shared__ float stage[NTH / 32][32 * 8];
  __shared__ float rbuf[NTH];
  load_WV(W, V, lW, lV);
  const int  idx   = blockIdx.x * NTH + threadIdx.x;
  const bool valid = idx < BS;
  const int  si    = valid ? idx : (BS - 1);
  __builtin_prefetch(&y0[(size_t)si * 8], 0, 3);
  const float dt = dtp[0], tt = tp[0];
  const float cdiag = 1.0f / dt + 0.1f;
  const int   isn = g->is_newton, rec = g->recompute;   // block-uniform flags
  const float lam = g->lam;
  float* stw = stage[threadIdx.x >> 5];

  float y[8], Fst[8];
#pragma unroll
  for (int k = 0; k < 8; k++) {
    y[k]   = out[(size_t)si * 8 + k];
    Fst[k] = out[(size_t)BS * NY + (size_t)si * 8 + k];
  }

  float su[8], dy[8], yo[8], foldv, slopev;
  float maxdyrel = -INFINITY;

  if (isn) {                                       // fresh Newton direction
    if (rec) {
      float u[8];
      wave_matvec8(stw, lW, y, u);                 // u = y @ W  (WMMA)
#pragma unroll
      for (int k = 0; k < 8; k++) { float th = tanhf(u[k]); su[k] = 1.0f - th * th; }
    } else {
#pragma unroll
      for (int k = 0; k < 8; k++) su[k] = wsf(ws, SB_O)[(size_t)si * 8 + k];
    }
    // J[i][j] = c*delta_ij - sum_k W[j,k]*s_k*V[k,i]
    float Jm[8][8];
#pragma unroll
    for (int i = 0; i < 8; i++)
#pragma unroll
      for (int j = 0; j < 8; j++) {
        float acc = 0.f;
#pragma unroll
        for (int k = 0; k < 8; k++) acc += lW[j * 16 + k] * su[k] * lV[k * 16 + i];
        Jm[i][j] = ((i == j) ? cdiag : 0.f) - acc;
      }
    float gv[8];
#pragma unroll
    for (int j = 0; j < 8; j++) {
      float acc = 0.f;
#pragma unroll
      for (int i = 0; i < 8; i++) acc += Fst[i] * Jm[i][j];
      gv[j] = acc;
    }
    float x[8];
#pragma unroll
    for (int i = 0; i < 8; i++) x[i] = -Fst[i];
    // pivot-free GE (J strongly diagonally dominant: diag~10.1, offdiag~0.08)
#pragma unroll
    for (int k = 0; k < 8; k++) {
      float inv = 1.0f / Jm[k][k];
#pragma unroll
      for (int j = 0; j < 8; j++) if (j > k) Jm[k][j] *= inv;
      x[k] *= inv;
#pragma unroll
      for (int i = 0; i < 8; i++) if (i > k) {
        float f = Jm[i][k];
#pragma unroll
        for (int j = 0; j < 8; j++) if (j > k) Jm[i][j] -= f * Jm[k][j];
        x[i] -= f * x[k];
      }
    }
#pragma unroll
    for (int k = 7; k >= 0; k--) {
      float acc = x[k];
#pragma unroll
      for (int j = 0; j < 8; j++) if (j > k) acc -= Jm[k][j] * x[j];
      x[k] = acc;
    }
    slopev = 0.f; foldv = 0.f;
#pragma unroll
    for (int k = 0; k < 8; k++) {
      dy[k] = x[k]; yo[k] = y[k];
      slopev += gv[k] * x[k];
      foldv  += Fst[k] * Fst[k];
      maxdyrel = fmaxf(maxdyrel, fabsf(x[k]) / fmaxf(fabsf(y[k]), 1.0f));
    }
  } else {                                         // backtracking retry: reuse direction
#pragma unroll
    for (int k = 0; k < 8; k++) {
      dy[k] = wsf(ws, DY_O)[(size_t)si * 8 + k];
      yo[k] = wsf(ws, YOLD_O)[(size_t)si * 8 + k];
      su[k] = wsf(ws, SB_O)[(size_t)si * 8 + k];
    }
    foldv  = wsf(ws, FOLD_O)[si];
    slopev = wsf(ws, SLOPE_O)[si];
  }

  // trial point + residual (two WMMA matvecs)
  float yn[8];
#pragma unroll
  for (int k = 0; k < 8; k++) yn[k] = yo[k] + lam * dy[k];
  float u2[8], th2[8], z2[8];
  wave_matvec8(stw, lW, yn, u2);
#pragma unroll
  for (int k = 0; k < 8; k++) th2[k] = tanhf(u2[k]);
  wave_matvec8(stw, lV, th2, z2);
  const float st = 0.01f * sinf(tt);
  float Fn[8], fobj = 0.f;
#pragma unroll
  for (int k = 0; k < 8; k++) {
    Fn[k] = (yn[k] - y0[(size_t)si * 8 + k]) / dt - z2[k] + 0.1f * yn[k] - st;
    fobj += Fn[k] * Fn[k];
  }

  // line-search candidates (quadratic + cubic models; branch selected globally)
  const float lam1v = lam, lam2v = g->lam2;
  const float f2v = wsf(ws, F2_O)[si];
  float C0 = fobj - foldv - lam1v * slopev;
  float C1 = f2v  - foldv - lam2v * slopev;
  float Ac = sdiv(1.0f, lam1v - lam2v);
  float i1 = sdiv(1.0f, lam1v * lam1v);
  float i2 = sdiv(1.0f, lam2v * lam2v);
  float a  = Ac * (i1 * C0 - i2 * C1);
  float b  = Ac * (-lam2v * i1 * C0 + lam1v * i2 * C1);
  float disc = b * b - 3.0f * a * slopev;
  float sq   = sqrtf(fmaxf(disc, 0.0f));
  float lamq  = sdiv(-0.5f * slopev, fobj - foldv - slopev);
  float cand1 = sdiv(-slopev, 2.0f * b);
  float cand3 = sdiv(-b + sq, 3.0f * a);
  float cand4 = sdiv(-slopev, b + sq);
  float btI  = (fobj > foldv + ALPHA_F * lam * slopev) ? 1.f : 0.f;
  float a0I  = (a == 0.0f) ? 1.f : 0.f;
  float dnI  = (disc < 0.0f) ? 1.f : 0.f;
  float bleI = (b <= 0.0f) ? 1.f : 0.f;
  float finDy = 1.f, finFo = finf(fobj);
#pragma unroll
  for (int k = 0; k < 8; k++) finDy *= finf(dy[k]);

  if (valid) {
#pragma unroll
    for (int k = 0; k < 8; k++) {
      wsf(ws, YTR_O)[(size_t)si * 8 + k]  = yn[k];
      wsf(ws, FTR_O)[(size_t)si * 8 + k]  = Fn[k];
      wsf(ws, DYTR_O)[(size_t)si * 8 + k] = dy[k];
      wsf(ws, SBTR_O)[(size_t)si * 8 + k] = su[k];
    }
    wsf(ws, SLTR_O)[si] = slopev;
  } else {  // neutral elements for reductions
    maxdyrel = -INFINITY; btI = 0.f;
    lamq = INFINITY; cand1 = INFINITY; cand3 = INFINITY; cand4 = INFINITY;
    a0I = 1.f; dnI = 0.f; bleI = 0.f; finDy = 1.f; finFo = 1.f;
  }
  float r[11];
  r[0]  = block_red(maxdyrel, 0, rbuf);
  r[1]  = block_red(btI,  0, rbuf);
  r[2]  = block_red(lamq, 1, rbuf);
  r[3]  = block_red(cand1,1, rbuf);
  r[4]  = block_red(cand3,1, rbuf);
  r[5]  = block_red(cand4,1, rbuf);
  r[6]  = block_red(a0I,  1, rbuf);
  r[7]  = block_red(dnI,  0, rbuf);
  r[8]  = block_red(bleI, 0, rbuf);
  r[9]  = block_red(finDy,1, rbuf);
  r[10] = block_red(finFo,1, rbuf);
  if (threadIdx.x == 0) {
    float* pa = wsf(ws, PA_OFF);
#pragma unroll
    for (int q = 0; q < 11; q++) pa[blockIdx.x * 12 + q] = r[q];
  }
}

__global__ __launch_bounds__(NTH) void k_reduceB(void* ws) {
  __shared__ float rbuf[NTH];
  Glb* g = (Glb*)ws;
  if (!g->active) {
    if (threadIdx.x == 0) { g->upd = 0; g->cont = 0; }
    return;
  }
  float* pa = wsf(ws, PA_OFF);
  const int isMin[11] = {0, 0, 1, 1, 1, 1, 1, 0, 0, 1, 1};
  float acc[11];
  for (int q = 0; q < 11; q++) acc[q] = isMin[q] ? INFINITY : -INFINITY;
  for (int i = threadIdx.x; i < NBLK; i += NTH) {
#pragma unroll
    for (int q = 0; q < 11; q++) {
      float v = pa[i * 12 + q];
      acc[q] = isMin[q] ? fminf(acc[q], v) : fmaxf(acc[q], v);
    }
  }
  float r[11];
  for (int q = 0; q < 11; q++) r[q] = block_red(acc[q], isMin[q], rbuf);
  if (threadIdx.x == 0) {
    const int   isn = g->is_newton;
    const float lam = g->lam;
    float lam_min = isn ? sdiv(TOLX_F, r[0]) : g->lam_min;
    int brk   = (lam < lam_min) ? 1 : 0;
    int bad   = (r[9]  < 0.5f) ? 1 : 0;
    int anyBT = (r[1]  > 0.5f) ? 1 : 0;
    int nanf  = (r[10] < 0.5f) ? 1 : 0;
    float lam1v = lam;
    float lam_tmp = (r[6] > 0.5f) ? r[3]
                  : ((r[7] > 0.5f) ? MAXLAM * lam1v
                  : ((r[8] > 0.5f) ? r[4] : r[5]));
    float lamc   = fminf(lam_tmp, MAXLAM * lam1v);
    float lam_bt = (lam == 1.0f) ? r[2] : lamc;
    float lam_new = anyBT ? lam_bt : (nanf ? MAXLAM * lam1v : 1.0f);
    int cont = (lam_new < 1.0f) ? 1 : 0;
    float lam2_new = cont ? lam1v : g->lam2;
    if (cont) lam_new = fmaxf(lam_new, MINLAM * lam1v);
    int upd = (!brk && !bad) ? 1 : 0;
    if (brk) g->exitflag = 2; else if (bad) g->exitflag = -1;
    if (isn) g->niter = g->niter + 1;
    if (upd) {
      g->lam = lam_new; g->lam1 = lam1v; g->lam2 = lam2_new;
      if (isn) g->lam_min = lam_min;
    }
    g->upd = upd; g->cont = cont;
  }
}

__global__ __launch_bounds__(NTH) void k_phaseC(float* out, void* ws) {
  __shared__ float rbuf[NTH];
  Glb* g = (Glb*)ws;
  const int  idx   = blockIdx.x * NTH + threadIdx.x;
  const bool valid = idx < BS;
  const int  si    = valid ? idx : (BS - 1);
  const int upd = g->upd, cont = g->cont, isn = g->is_newton;   // block-uniform
  float rn = wsf(ws, RN_O)[si], rn0 = wsf(ws, RN0_O)[si];
  if (upd) {
    float ytr[8], ftr[8], fobj = 0.f, rnew = 0.f;
#pragma unroll
    for (int k = 0; k < 8; k++) {
      ytr[k] = wsf(ws, YTR_O)[(size_t)si * 8 + k];
      ftr[k] = wsf(ws, FTR_O)[(size_t)si * 8 + k];
      fobj += ftr[k] * ftr[k];
      rnew = fmaxf(rnew, fabsf(ftr[k]));
    }
    if (valid) {
      if (isn) {   // commit Newton-path state: yold, dy, jac-basis, fold, slope
        float foldn = 0.f;
#pragma unroll
        for (int k = 0; k < 8; k++) {
          float yc = out[(size_t)si * 8 + k];
          float fc = out[(size_t)BS * NY + (size_t)si * 8 + k];
          wsf(ws, YOLD_O)[(size_t)si * 8 + k] = yc;
          wsf(ws, DY_O)[(size_t)si * 8 + k]   = wsf(ws, DYTR_O)[(size_t)si * 8 + k];
          wsf(ws, SB_O)[(size_t)si * 8 + k]   = wsf(ws, SBTR_O)[(size_t)si * 8 + k];
          foldn += fc * fc;
        }
        wsf(ws, FOLD_O)[si]  = foldn;
        wsf(ws, SLOPE_O)[si] = wsf(ws, SLTR_O)[si];
      }
      if (cont) {
        wsf(ws, F2_O)[si] = fobj;
      } else {
        rn0 = rn; rn = rnew;
        wsf(ws, RN_O)[si] = rn; wsf(ws, RN0_O)[si] = rn0;
      }
#pragma unroll
      for (int k = 0; k < 8; k++) {
        out[(size_t)si * 8 + k]                   = ytr[k];
        out[(size_t)BS * NY + (size_t)si * 8 + k] = ftr[k];
      }
    }
  }
  float ratio = sdiv(rn, rn0);
  float r0 = block_red(valid ? rn : -INFINITY, 0, rbuf);
  float r1 = block_red(valid ? ratio : -INFINITY, 0, rbuf);
  if (threadIdx.x == 0) {
    wsf(ws, PR_OFF)[blockIdx.x * 2 + 0] = r0;
    wsf(ws, PR_OFF)[blockIdx.x * 2 + 1] = r1;
  }
}

__global__ void k_final(float* out, void* ws) {
  Glb* g = (Glb*)ws;
  ((int*)out)[(size_t)BS * NY * 2] = g->exitflag;
}

// ---------------- entry point ----------------
extern "C" void kernel_launch(void* const* d_in, const int* in_sizes, int n_in,
                              void* d_out, int out_size, void* d_ws, size_t ws_size,
                              hipStream_t stream) {
  (void)in_sizes; (void)n_in; (void)out_size; (void)ws_size;
  const float* y0 = (const float*)d_in[0];
  const float* dt = (const float*)d_in[1];
  const float* t  = (const float*)d_in[2];
  const float* W  = (const float*)d_in[3];
  const float* V  = (const float*)d_in[4];
  float* out = (float*)d_out;
  dim3 blk(NTH), grd(NBLK);
  k_init_glb<<<1, 1, 0, stream>>>(d_ws);
  k_init<<<grd, blk, 0, stream>>>(y0, W, V, t, out, d_ws);
  for (int s = 0; s < N_STEPS; s++) {
    k_reduceR<<<1, NTH, 0, stream>>>(d_ws);
    k_phaseA<<<grd, blk, 0, stream>>>(y0, W, V, dt, t, out, d_ws);
    k_reduceB<<<1, NTH, 0, stream>>>(d_ws);
    k_phaseC<<<grd, blk, 0, stream>>>(out, d_ws);
  }
  k_final<<<1, 1, 0, stream>>>(out, d_ws);
}